// Metalayer_81200651698402
// MI455X (gfx1250) — compile-verified
//
#include <hip/hip_runtime.h>
#include <math.h>
#include <utility>

#define NWG   512
#define RESN  8192
#define LWAV  3
#define HID   64
#define PERIODF 0.3f
#define WHF     0.6f
#define MATN  512
#define MAT_ELEMS (MATN*MATN)
#define EXPM_S   7
#define EXPM_DEG 10
#define NPAIR 2042

typedef __attribute__((ext_vector_type(2))) float v2f;
typedef __attribute__((ext_vector_type(8))) float v8f;

__device__ __forceinline__ v8f wmma4(v2f a, v2f b, v8f c) {
  // V_WMMA_F32_16X16X4_F32 : D = A(16x4) x B(4x16) + C(16x16)
  return __builtin_amdgcn_wmma_f32_16x16x4_f32(false, a, false, b, (short)0, c,
                                               false, false);
}

__device__ __forceinline__ float waveRedSum(float v) {
  #pragma unroll
  for (int o = 16; o > 0; o >>= 1) v += __shfl_xor(v, o, 32);
  return v;
}

// ---------------- small elementwise / MLP kernels ----------------

__global__ void hs_kernel(const float* hp, float* hs) {
  int i = blockIdx.x * blockDim.x + threadIdx.x;
  if (i < NWG) {
    float s = 1.0f / (1.0f + expf(-hp[i]));
    hs[i] = s * 0.2f + 0.1f;                       // H_MIN + s*(H_MAX-H_MIN)
  }
}

__global__ void neff_kernel(const float* hs, const float* nW1, const float* nb1,
                            const float* nW2, const float* nb2,
                            float* neff, float* rsq, int l) {
  int n = blockIdx.x;            // 512 blocks x 64 threads
  int h = threadIdx.x;
  __shared__ float sh[HID];
  float x = hs[n];
  float z = tanhf(x * nW1[l * HID + h] + nb1[l * HID + h]);
  sh[h] = z * nW2[l * HID + h];
  __syncthreads();
  if (h == 0) {
    float acc = nb2[l];
    for (int i = 0; i < HID; ++i) acc += sh[i];
    float ne = 1.0f + 2.0f / (1.0f + expf(-acc));  // neff in (1,3)
    neff[n] = ne;
    rsq[n] = rsqrtf(ne);
  }
}

__global__ void hidden_kernel(const float* hs, const float* eW1, const float* eb1,
                              float* H, int l) {
  int idx = blockIdx.x * blockDim.x + threadIdx.x;  // 512*64
  int n = idx >> 6, h = idx & 63;
  H[idx] = tanhf(hs[n] * eW1[l * HID + h] + eb1[l * HID + h]);
}

// ---------------- G = E0g(512x8192) @ eW2^T(8192x64)  [WMMA f32] ----------------

__global__ void gemmG_kernel(const float* __restrict__ E0,
                             const float* __restrict__ eW2,
                             float* __restrict__ G, int l) {
  int gw = (blockIdx.x * blockDim.x + threadIdx.x) >> 5;  // 128 waves exactly
  int lane = threadIdx.x & 31;
  int tm = gw >> 2;            // 32 row tiles
  int tn = gw & 3;             // 4 col tiles
  int cc = lane & 15, hh = lane >> 4;
  int r0 = tm * 16, c0 = tn * 16;
  const float* W = eW2 + (size_t)l * HID * RESN;
  const float* arow = E0 + (size_t)(r0 + cc) * RESN;
  const float* brow = W + (size_t)(c0 + cc) * RESN;   // B[k][j] = eW2[j][k]
  v8f acc = {};
  for (int k = 0; k < RESN; k += 4) {
    int k0 = k + 2 * hh;
    v2f a, b;
    a.x = arow[k0]; a.y = arow[k0 + 1];
    b.x = brow[k0]; b.y = brow[k0 + 1];
    acc = wmma4(a, b, acc);
  }
  #pragma unroll
  for (int g = 0; g < 8; ++g) {
    int row = r0 + g + 8 * hh, col = c0 + cc;
    G[row * HID + col] = acc[g];
  }
}

// U0[n] = (sum_r E0[n,r]*eb2[r] + H[n,:].G[n,:]) * 2*DX
__global__ void u0_kernel(const float* E0, const float* eb2, const float* H,
                          const float* G, float* U0, int l) {
  int n = blockIdx.x;            // 512 blocks x 32 threads
  int lane = threadIdx.x;
  const float* row = E0 + (size_t)n * RESN;
  const float* bias = eb2 + (size_t)l * RESN;
  float acc = 0.f;
  for (int r = lane; r < RESN; r += 32) {
    if (r + 1024 < RESN) __builtin_prefetch(row + r + 1024, 0, 1);
    acc += row[r] * bias[r];
  }
  acc = waveRedSum(acc);
  if (lane == 0) {
    float dot = 0.f;
    for (int h = 0; h < HID; ++h) dot += H[n * HID + h] * G[n * HID + h];
    U0[n] = (acc + dot) * (2.0f * (PERIODF / RESN));
  }
}

// ---------------- coupling matrices ----------------

__global__ void initCK_kernel(float* C, float* M) {
  int idx = blockIdx.x * blockDim.x + threadIdx.x;
  int i = idx >> 9, j = idx & 511;
  C[idx] = (i == j) ? 1.0f : 0.0f;
  M[idx] = 0.0f;
}

__global__ void coupling_kernel(const float* hs,
                                const float* cW1, const float* cb1,
                                const float* cW2, const float* cb2,
                                const float* kW1, const float* kb1,
                                const float* kW2, const float* kb2,
                                float* C, float* M, int l) {
  int p = blockIdx.x * blockDim.x + threadIdx.x;
  if (p >= NPAIR) return;
  int i, j; float d;
  if (p < 510)       { i = 2 + p;           j = i - 2; d = 2.0f * PERIODF; }
  else if (p < 1021) { i = 1 + (p - 510);   j = i - 1; d = PERIODF; }
  else if (p < 1532) { i = (p - 1021);      j = i + 1; d = PERIODF; }
  else               { i = (p - 1532);      j = i + 2; d = 2.0f * PERIODF; }
  float x0 = hs[i], x1 = hs[j];
  const float* cw = cW1 + l * 3 * HID;
  const float* kw = kW1 + l * 3 * HID;
  float ca = 0.f, ka = 0.f;
  for (int h = 0; h < HID; ++h) {
    float zc = tanhf(x0 * cw[h] + x1 * cw[HID + h] + d * cw[2 * HID + h] +
                     cb1[l * HID + h]);
    ca += zc * cW2[l * HID + h];
    float zk = tanhf(x0 * kw[h] + x1 * kw[HID + h] + d * kw[2 * HID + h] +
                     kb1[l * HID + h]);
    ka += zk * kW2[l * HID + h];
  }
  C[i * MATN + j] = ca + cb2[l];
  M[i * MATN + j] = ka + kb2[l];
}

// M = diag(neff*k) * C + K   (K already in M)
__global__ void formM_kernel(const float* C, const float* neff, float* M,
                             float kwav) {
  int idx = blockIdx.x * blockDim.x + threadIdx.x;
  int i = idx >> 9;
  M[idx] += neff[i] * kwav * C[idx];
}

// ---------------- banded solve: C X = M, C pentadiagonal ----------------

__global__ void bandsolve_kernel(const float* C, float* M) {
  __shared__ float am2[MATN], am1[MATN], a0[MATN], ap1[MATN], ap2[MATN];
  int t = threadIdx.x;           // 512 threads, 1 block
  am2[t] = (t >= 2)   ? C[t * MATN + t - 2] : 0.f;
  am1[t] = (t >= 1)   ? C[t * MATN + t - 1] : 0.f;
  a0[t]  =              C[t * MATN + t];
  ap1[t] = (t < 511)  ? C[t * MATN + t + 1] : 0.f;
  ap2[t] = (t < 510)  ? C[t * MATN + t + 2] : 0.f;
  __syncthreads();
  if (t == 0) {                  // banded LU, no pivoting (diag ~ 1)
    for (int r = 0; r < MATN; ++r) {
      float piv = a0[r];
      if (r + 1 < MATN) {
        float m1 = am1[r + 1] / piv; am1[r + 1] = m1;
        a0[r + 1]  -= m1 * ap1[r];
        ap1[r + 1] -= m1 * ap2[r];
      }
      if (r + 2 < MATN) {
        float m2 = am2[r + 2] / piv; am2[r + 2] = m2;
        am1[r + 2] -= m2 * ap1[r];
        a0[r + 2]  -= m2 * ap2[r];
      }
    }
  }
  __syncthreads();
  // each thread solves one RHS column in place
  float ym1 = 0.f, ym2 = 0.f;
  for (int r = 0; r < MATN; ++r) {
    float v = M[r * MATN + t] - am1[r] * ym1 - am2[r] * ym2;
    M[r * MATN + t] = v; ym2 = ym1; ym1 = v;
  }
  float xp1 = 0.f, xp2 = 0.f;
  for (int r = MATN - 1; r >= 0; --r) {
    float v = (M[r * MATN + t] - ap1[r] * xp1 - ap2[r] * xp2) / a0[r];
    M[r * MATN + t] = v; xp2 = xp1; xp1 = v;
  }
}

// EigM = mu*I + R (exact split; scalar phase e^{i*WH*mu} folded into amp).
// B = theta*R with theta = WH/2^s ; exp(iB) = cos(B) + i sin(B):
//   Sre = I (k=0 term), Sim = B (k=1 term), T = B
__global__ void makeB_kernel(const float* EigM, float* B,
                             float* Sre, float* Sim, float* T,
                             float theta, float mu) {
  int idx = blockIdx.x * blockDim.x + threadIdx.x;
  int i = idx >> 9, j = idx & 511;
  float diag = (i == j) ? 1.0f : 0.0f;
  float bv = theta * (EigM[idx] - mu * diag);
  B[idx] = bv;
  Sre[idx] = diag;
  Sim[idx] = bv;
  T[idx] = bv;
}

// ---------------- real 512x512x512 GEMM on f32 WMMA (Taylor terms) ------------

__global__ void rgemm_kernel(const float* __restrict__ X,
                             const float* __restrict__ Y,
                             float* __restrict__ W, float alpha) {
  int gw = (blockIdx.x * blockDim.x + threadIdx.x) >> 5;  // 1024 waves exactly
  int lane = threadIdx.x & 31;
  int tm = gw >> 5, tn = gw & 31;
  int cc = lane & 15, hh = lane >> 4;
  int r0 = tm * 16, c0 = tn * 16;
  v8f acc = {};
  const float* xr = X + (size_t)(r0 + cc) * MATN;
  for (int k = 0; k < MATN; k += 4) {
    int k0 = k + 2 * hh;
    v2f a, b;
    a.x = xr[k0];  a.y = xr[k0 + 1];
    b.x = Y[(size_t)k0 * MATN + c0 + cc];
    b.y = Y[(size_t)(k0 + 1) * MATN + c0 + cc];
    acc = wmma4(a, b, acc);
  }
  #pragma unroll
  for (int g = 0; g < 8; ++g) {
    int row = r0 + g + 8 * hh, col = c0 + cc;
    W[row * MATN + col] = alpha * acc[g];
  }
}

// dst += sign * T   (accumulate Taylor term into cos or sin plane)
__global__ void addsign_kernel(float* dst, const float* T, float sign) {
  int idx = blockIdx.x * blockDim.x + threadIdx.x;
  dst[idx] += sign * T[idx];
}

// ---------------- complex 512x512 GEMM on f32 WMMA (squaring) ----------------

__global__ void cgemm_kernel(const float* __restrict__ Xre, const float* __restrict__ Xim,
                             const float* __restrict__ Yre, const float* __restrict__ Yim,
                             float* __restrict__ Wre, float* __restrict__ Wim) {
  int gw = (blockIdx.x * blockDim.x + threadIdx.x) >> 5;  // 1024 waves exactly
  int lane = threadIdx.x & 31;
  int tm = gw >> 5, tn = gw & 31;
  int cc = lane & 15, hh = lane >> 4;
  int r0 = tm * 16, c0 = tn * 16;
  v8f aRR = {}, aII = {}, aRI = {}, aIR = {};
  const float* xr = Xre + (size_t)(r0 + cc) * MATN;
  const float* xi = Xim + (size_t)(r0 + cc) * MATN;
  for (int k = 0; k < MATN; k += 4) {
    int k0 = k + 2 * hh;
    v2f ar, ai, br, bi;
    ar.x = xr[k0];  ar.y = xr[k0 + 1];
    ai.x = xi[k0];  ai.y = xi[k0 + 1];
    br.x = Yre[(size_t)k0 * MATN + c0 + cc];
    br.y = Yre[(size_t)(k0 + 1) * MATN + c0 + cc];
    bi.x = Yim[(size_t)k0 * MATN + c0 + cc];
    bi.y = Yim[(size_t)(k0 + 1) * MATN + c0 + cc];
    aRR = wmma4(ar, br, aRR);
    aII = wmma4(ai, bi, aII);
    aRI = wmma4(ar, bi, aRI);
    aIR = wmma4(ai, br, aIR);
  }
  #pragma unroll
  for (int g = 0; g < 8; ++g) {
    int row = r0 + g + 8 * hh, col = c0 + cc;
    Wre[row * MATN + col] = aRR[g] - aII[g];
    Wim[row * MATN + col] = aRI[g] + aIR[g];
  }
}

// amp = e^{i*phi} * (P @ U0) / sqrt(neff)
__global__ void matvec_kernel(const float* Pre, const float* Pim,
                              const float* U0, const float* rsq,
                              float* ampRe, float* ampIm,
                              float cphi, float sphi) {
  int n = blockIdx.x;            // 512 blocks x 32 threads
  int lane = threadIdx.x;
  float ar = 0.f, ai = 0.f;
  for (int j = lane; j < MATN; j += 32) {
    float u = U0[j];
    ar += Pre[n * MATN + j] * u;
    ai += Pim[n * MATN + j] * u;
  }
  ar = waveRedSum(ar);
  ai = waveRedSum(ai);
  if (lane == 0) {
    float s = rsq[n];
    ampRe[n] = (ar * cphi - ai * sphi) * s;
    ampIm[n] = (ar * sphi + ai * cphi) * s;
  }
}

// En = amp ⊗ Eys,  Eys = H(512x64) @ eW2(64x8192) + eb2   [WMMA f32, fused epilogue]
__global__ void gemmOut_kernel(const float* __restrict__ H,
                               const float* __restrict__ eW2,
                               const float* __restrict__ eb2,
                               const float* __restrict__ ampRe,
                               const float* __restrict__ ampIm,
                               float* __restrict__ out, int l) {
  int gw = (blockIdx.x * blockDim.x + threadIdx.x) >> 5;  // 16384 waves exactly
  int lane = threadIdx.x & 31;
  int tm = gw >> 9;              // 32 row tiles
  int tn = gw & 511;             // 512 col tiles
  int cc = lane & 15, hh = lane >> 4;
  int r0 = tm * 16, c0 = tn * 16;
  const float* W = eW2 + (size_t)l * HID * RESN;
  const float* bias = eb2 + (size_t)l * RESN;
  const float* arow = H + (size_t)(r0 + cc) * HID;
  v8f acc = {};
  #pragma unroll
  for (int k = 0; k < HID; k += 4) {
    int k0 = k + 2 * hh;
    v2f a, b;
    a.x = arow[k0];  a.y = arow[k0 + 1];
    b.x = W[(size_t)k0 * RESN + c0 + cc];
    b.y = W[(size_t)(k0 + 1) * RESN + c0 + cc];
    acc = wmma4(a, b, acc);
  }
  float* dst = out + (size_t)l * (2ull * NWG * RESN);
  #pragma unroll
  for (int g = 0; g < 8; ++g) {
    int row = r0 + g + 8 * hh, col = c0 + cc;
    float e = acc[g] + bias[col];
    size_t o = 2ull * ((size_t)row * RESN + col);
    dst[o]     = ampRe[row] * e;
    dst[o + 1] = ampIm[row] * e;
  }
}

// ---------------- host orchestration ----------------

extern "C" void kernel_launch(void* const* d_in, const int* in_sizes, int n_in,
                              void* d_out, int out_size, void* d_ws, size_t ws_size,
                              hipStream_t stream) {
  (void)in_sizes; (void)n_in; (void)out_size; (void)ws_size;
  const float* E0  = (const float*)d_in[0];
  const float* hp  = (const float*)d_in[1];
  const float* nW1 = (const float*)d_in[2];
  const float* nb1 = (const float*)d_in[3];
  const float* nW2 = (const float*)d_in[4];
  const float* nb2 = (const float*)d_in[5];
  const float* cW1 = (const float*)d_in[6];
  const float* cb1 = (const float*)d_in[7];
  const float* cW2 = (const float*)d_in[8];
  const float* cb2 = (const float*)d_in[9];
  const float* kW1 = (const float*)d_in[10];
  const float* kb1 = (const float*)d_in[11];
  const float* kW2 = (const float*)d_in[12];
  const float* kb2 = (const float*)d_in[13];
  const float* eW1 = (const float*)d_in[14];
  const float* eb1 = (const float*)d_in[15];
  const float* eW2 = (const float*)d_in[16];
  const float* eb2 = (const float*)d_in[17];
  float* out = (float*)d_out;
  float* ws  = (float*)d_ws;

  float* hs    = ws;                 // 512
  float* neff  = ws + 512;
  float* rsq   = ws + 1024;
  float* U0    = ws + 1536;
  float* ampRe = ws + 2048;
  float* ampIm = ws + 2560;
  float* H     = ws + 3072;          // 512*64
  float* G     = ws + 3072 + NWG * HID;
  float* base  = ws + 3072 + 2 * NWG * HID;   // 9 x 512x512 matrices
  float* Cm   = base + 0 * MAT_ELEMS;
  float* Mm   = base + 1 * MAT_ELEMS;
  float* Bs   = base + 2 * MAT_ELEMS;
  float* T0   = base + 3 * MAT_ELEMS;
  float* T1   = base + 4 * MAT_ELEMS;
  float* S0re = base + 5 * MAT_ELEMS;
  float* S0im = base + 6 * MAT_ELEMS;
  float* S1re = base + 7 * MAT_ELEMS;
  float* S1im = base + 8 * MAT_ELEMS;

  hs_kernel<<<2, 256, 0, stream>>>(hp, hs);

  const float lam[LWAV] = {0.45f, 0.55f, 0.65f};
  for (int l = 0; l < LWAV; ++l) {
    float kwav = 6.283185307179586f / lam[l];

    neff_kernel<<<NWG, HID, 0, stream>>>(hs, nW1, nb1, nW2, nb2, neff, rsq, l);
    hidden_kernel<<<(NWG * HID) / 256, 256, 0, stream>>>(hs, eW1, eb1, H, l);
    gemmG_kernel<<<16, 256, 0, stream>>>(E0, eW2, G, l);
    u0_kernel<<<NWG, 32, 0, stream>>>(E0, eb2, H, G, U0, l);

    initCK_kernel<<<MAT_ELEMS / 256, 256, 0, stream>>>(Cm, Mm);
    coupling_kernel<<<(NPAIR + 255) / 256, 256, 0, stream>>>(
        hs, cW1, cb1, cW2, cb2, kW1, kb1, kW2, kb2, Cm, Mm, l);
    formM_kernel<<<MAT_ELEMS / 256, 256, 0, stream>>>(Cm, neff, Mm, kwav);
    bandsolve_kernel<<<1, MATN, 0, stream>>>(Cm, Mm);   // Mm := EigM

    // exp(i*WH*EigM) = e^{i*WH*mu} * exp(i*WH*R),  R = EigM - mu*I (exact)
    float mu = 2.0f * kwav;                 // center of neff range (1,3)
    float theta = WHF / (float)(1 << EXPM_S);
    makeB_kernel<<<MAT_ELEMS / 256, 256, 0, stream>>>(Mm, Bs, S0re, S0im, T0,
                                                      theta, mu);
    // Taylor: exp(iB) = cos(B) + i sin(B) using REAL powers B^k/k!
    float *Sre = S0re, *Sim = S0im, *Qre = S1re, *Qim = S1im;
    float *Tc = T0, *Tn = T1;
    for (int kk = 2; kk <= EXPM_DEG; ++kk) {
      rgemm_kernel<<<128, 256, 0, stream>>>(Tc, Bs, Tn, 1.0f / (float)kk);
      int m4 = kk & 3;                       // i^k sign pattern
      float sign = (m4 == 0 || m4 == 1) ? 1.0f : -1.0f;
      float* target = (kk & 1) ? Sim : Sre;
      addsign_kernel<<<MAT_ELEMS / 256, 256, 0, stream>>>(target, Tn, sign);
      std::swap(Tc, Tn);
    }
    // repeated squaring (complex)
    for (int sq = 0; sq < EXPM_S; ++sq) {
      cgemm_kernel<<<128, 256, 0, stream>>>(Sre, Sim, Sre, Sim, Qre, Qim);
      std::swap(Sre, Qre); std::swap(Sim, Qim);
    }

    float phi = WHF * mu;
    matvec_kernel<<<NWG, 32, 0, stream>>>(Sre, Sim, U0, rsq, ampRe, ampIm,
                                          cosf(phi), sinf(phi));
    gemmOut_kernel<<<2048, 256, 0, stream>>>(H, eW2, eb2, ampRe, ampIm, out, l);
  }
}